// PGDVSBaseRenderer_70660801954136
// MI455X (gfx1250) — compile-verified
//
#include <hip/hip_runtime.h>
#include <cstdint>

namespace {
constexpr int B_ = 4, C_ = 3, H_ = 1024, W_ = 1920;
constexpr int HW   = H_ * W_;        // 1,966,080
constexpr int NPIX = B_ * HW;        // 7,864,320
constexpr float ALPHA = 100.0f;
constexpr float EPS   = 1e-7f;
}

typedef unsigned int v4u __attribute__((ext_vector_type(4)));
typedef unsigned int v8u __attribute__((ext_vector_type(8)));

// Fused: backwarp(rgb2, flow12) -> metric -> em -> forward-splat scatter-add.
// One thread per source pixel. Planar layout keeps every streaming load coalesced.
__global__ __launch_bounds__(256) void softsplat_fused(
    const float* __restrict__ rgb1,
    const float* __restrict__ rgb2,
    const float* __restrict__ flow_t,    // flow_src1_to_tgt  [B,2,H,W]
    const float* __restrict__ flow_12,   // flow_src1_to_src2 [B,2,H,W]
    float* __restrict__ acc_rgb,         // [B,3,H,W] (zero-initialized, lives in d_out)
    float* __restrict__ acc_w,           // [B,H,W]   (zero-initialized, lives in d_ws)
    float* __restrict__ metric_out)      // [B,H,W]
{
    int i = blockIdx.x * blockDim.x + threadIdx.x;
    if (i >= NPIX) return;
    int b = i / HW;
    int r = i - b * HW;
    int y = r / W_;
    int x = r - y * W_;

    // ------- backward warp of rgb2 at p + flow12 (bilinear, zero padding) -------
    const float* f12 = flow_12 + b * 2 * HW + r;
    float px = (float)x + f12[0];
    float py = (float)y + f12[HW];
    float x0f = floorf(px), y0f = floorf(py);
    int   x0  = (int)x0f,   y0  = (int)y0f;
    float wx  = px - x0f,   wy  = py - y0f;

    bool vx0 = (unsigned)x0       < (unsigned)W_;
    bool vx1 = (unsigned)(x0 + 1) < (unsigned)W_;
    bool vy0 = (unsigned)y0       < (unsigned)H_;
    bool vy1 = (unsigned)(y0 + 1) < (unsigned)H_;
    int cx0 = min(max(x0, 0), W_ - 1);
    int cx1 = min(max(x0 + 1, 0), W_ - 1);
    int cy0 = min(max(y0, 0), H_ - 1);
    int cy1 = min(max(y0 + 1, 0), H_ - 1);

    float m00 = (vx0 && vy0) ? (1.f - wx) * (1.f - wy) : 0.f;
    float m10 = (vx1 && vy0) ? wx * (1.f - wy)         : 0.f;
    float m01 = (vx0 && vy1) ? (1.f - wx) * wy         : 0.f;
    float m11 = (vx1 && vy1) ? wx * wy                 : 0.f;

    const float* p2 = rgb2 + b * C_ * HW;
    int i00 = cy0 * W_ + cx0, i10 = cy0 * W_ + cx1;
    int i01 = cy1 * W_ + cx0, i11 = cy1 * W_ + cx1;

    const float* p1 = rgb1 + b * C_ * HW + r;
    float s0 = p1[0], s1 = p1[HW], s2 = p1[2 * HW];

    float md;
    {
        float g0 = m00 * p2[i00] + m10 * p2[i10] + m01 * p2[i01] + m11 * p2[i11];
        const float* q = p2 + HW;
        float g1 = m00 * q[i00] + m10 * q[i10] + m01 * q[i01] + m11 * q[i11];
        q += HW;
        float g2 = m00 * q[i00] + m10 * q[i10] + m01 * q[i01] + m11 * q[i11];
        md = fabsf(s0 - g0) + fabsf(s1 - g1) + fabsf(s2 - g2);
    }
    float metric = md * (1.f / 3.f);
    metric_out[i] = metric;

    float m  = fminf(fmaxf(-ALPHA * metric, -ALPHA), ALPHA);
    float em = expf(m);

    // ------- forward splat (scatter-add) at p + flow_tgt -------
    const float* ft = flow_t + b * 2 * HW + r;
    float tx = (float)x + ft[0];
    float ty = (float)y + ft[HW];
    float tx0f = floorf(tx), ty0f = floorf(ty);
    int   tx0  = (int)tx0f,  ty0  = (int)ty0f;
    float twx  = tx - tx0f,  twy  = ty - ty0f;

    float v0 = s0 * em, v1 = s1 * em, v2 = s2 * em;
    float* ob = acc_rgb + b * C_ * HW;
    float* wb = acc_w   + b * HW;

    #pragma unroll
    for (int k = 0; k < 4; ++k) {
        int   xi = tx0 + (k & 1);
        int   yi = ty0 + (k >> 1);
        float w  = ((k & 1) ? twx : 1.f - twx) * ((k >> 1) ? twy : 1.f - twy);
        if ((unsigned)xi < (unsigned)W_ && (unsigned)yi < (unsigned)H_) {
            int o = yi * W_ + xi;
            // Hardware fp32 fadd atomics (non-returning -> STOREcnt path).
            unsafeAtomicAdd(ob + o,          v0 * w);
            unsafeAtomicAdd(ob + HW + o,     v1 * w);
            unsafeAtomicAdd(ob + 2 * HW + o, v2 * w);
            unsafeAtomicAdd(wb + o,          em * w);
        }
    }
}

// In-place normalize: out_rgb /= (acc_w + EPS).
// The weight plane (the normalization divisor) is staged into LDS by the Tensor
// Data Mover: each wave issues one 512B tensor_load_to_lds, overlaps the DMA with
// its three coalesced B128 rgb-plane loads, then waits on its own TENSORcnt.
// Grid is exact: (NPIX/4) threads, 256/block -> no bounds checks, EXEC all ones.
__global__ __launch_bounds__(256) void softsplat_norm(
    float* __restrict__ out, const float* __restrict__ acc_w)
{
    __shared__ __align__(16) float lds_w[1024];  // 4 KB weight tile (TDM destination)

    int tid  = threadIdx.x;
    int wave = tid >> 5;                     // 8 waves per block
    int lane = tid & 31;
    int j    = blockIdx.x * 256 + tid;       // float4 index over B*HW/4

    // ---- Tensor DMA descriptor (D#), per CDNA5 ISA ch.8: 1-row 2D tile ----
    uint64_t gaddr = (uint64_t)(uintptr_t)acc_w
                   + (uint64_t)blockIdx.x * 4096u + (uint64_t)wave * 512u;
    uint32_t lds_off = (uint32_t)wave * 512u;     // lds_w sits at LDS offset 0

    v4u g0;
    g0.x = 1u;                                       // count=1, user mode
    g0.y = lds_off;                                  // lds_addr (bytes)
    g0.z = (uint32_t)gaddr;                          // global_addr[31:0]
    g0.w = ((uint32_t)(gaddr >> 32) & 0x01FFFFFFu)   // global_addr[56:32]
         | (2u << 30);                               // type = 2

    v8u g1;
    g1.s0 = 2u << 16;       // workgroup_mask=0, data_size=2 (4B), no iterate/pad
    g1.s1 = 128u << 16;     // tensor_dim0 = 128 (low 16 bits -> [31:16])
    g1.s2 = 1u << 16;       // tensor_dim0 hi = 0 | tensor_dim1 = 1
    g1.s3 = 128u << 16;     // tensor_dim1 hi = 0 | tile_dim0 = 128
    g1.s4 = 1u;             // tile_dim1 = 1, tile_dim2 = 0
    g1.s5 = 128u;           // tensor_dim0_stride = 128 (low 32)
    g1.s6 = 0u;
    g1.s7 = 0u;

    // The LDS pointer is passed as an operand so the shared array escapes into
    // the asm: combined with the "memory" clobber this forces the compiler to
    // treat lds_w as written by the TDM (otherwise the ds_load below is folded
    // to poison, as observed in the round-2 histogram: ds=0).
    asm volatile("tensor_load_to_lds %0, %1"
                 :
                 : "s"(g0), "s"(g1), "v"(&lds_w[0])
                 : "memory");

    // ---- overlap the tensor DMA with the three rgb-plane B128 loads ----
    int b  = j / (HW / 4);
    int r4 = j - b * (HW / 4);
    float4* ob = reinterpret_cast<float4*>(out) + b * (C_ * HW / 4) + r4;
    float4 v0 = ob[0];
    float4 v1 = ob[HW / 4];
    float4 v2 = ob[2 * (HW / 4)];

    __builtin_amdgcn_s_wait_tensorcnt(0);
    asm volatile("" ::: "memory");

    const float4 wv =
        *reinterpret_cast<const float4*>(&lds_w[wave * 128 + lane * 4]);
    float4 rr;
    rr.x = 1.f / (wv.x + EPS);
    rr.y = 1.f / (wv.y + EPS);
    rr.z = 1.f / (wv.z + EPS);
    rr.w = 1.f / (wv.w + EPS);

    v0.x *= rr.x; v0.y *= rr.y; v0.z *= rr.z; v0.w *= rr.w;
    v1.x *= rr.x; v1.y *= rr.y; v1.z *= rr.z; v1.w *= rr.w;
    v2.x *= rr.x; v2.y *= rr.y; v2.z *= rr.z; v2.w *= rr.w;

    ob[0]               = v0;
    ob[HW / 4]          = v1;
    ob[2 * (HW / 4)]    = v2;
}

extern "C" void kernel_launch(void* const* d_in, const int* in_sizes, int n_in,
                              void* d_out, int out_size, void* d_ws, size_t ws_size,
                              hipStream_t stream) {
    const float* rgb1    = (const float*)d_in[0];
    const float* rgb2    = (const float*)d_in[1];
    const float* flow_t  = (const float*)d_in[2];
    const float* flow_12 = (const float*)d_in[3];

    float* out        = (float*)d_out;                    // [B,3,H,W] splat_img, then [B,1,H,W] metric
    float* metric_out = out + (size_t)B_ * C_ * HW;
    float* acc_w      = (float*)d_ws;                     // [B,H,W] weight accumulator

    // Accumulators must be zero at the start of every call (graph replays included).
    hipMemsetAsync(out,   0, (size_t)B_ * C_ * HW * sizeof(float), stream);
    hipMemsetAsync(acc_w, 0, (size_t)B_ * HW * sizeof(float), stream);

    softsplat_fused<<<(NPIX + 255) / 256, 256, 0, stream>>>(
        rgb1, rgb2, flow_t, flow_12, out, acc_w, metric_out);

    softsplat_norm<<<(NPIX / 4) / 256, 256, 0, stream>>>(out, acc_w);
}